// MPSDist_54236847014559
// MI455X (gfx1250) — compile-verified
//
#include <hip/hip_runtime.h>
#include <cstdint>
#include <cstddef>

// Problem constants (match reference)
#define NV    32000   // vocab
#define RR    8       // rank
#define BB    32      // batch
#define TT    64      // seq
#define NROWS (BB*TT) // 2048 output rows
#define KDIM  (RR*RR) // 64 = GEMM K
#define NCB   (NV/128) // 250 column blocks in k_gemm

typedef __attribute__((ext_vector_type(2))) float v2f;
typedef __attribute__((ext_vector_type(4))) float v4f;
typedef __attribute__((ext_vector_type(8))) float v8f;

// ---------------------------------------------------------------------------
// Kernel 1a: partial vocab-marginalization of core^2 -> 64 partial 8x8 blocks.
// Deterministic (no float atomics): block b reduces v in [b*500, b*500+500).
// ---------------------------------------------------------------------------
__global__ void __launch_bounds__(256)
k_marg_partial(const float* __restrict__ core, float* __restrict__ mpart) {
    __shared__ float red[256];
    const int tid   = threadIdx.x;
    const int pair  = tid & 63;       // i*8 + j
    const int slice = tid >> 6;       // 0..3
    const int i = pair >> 3, j = pair & 7;
    const int v0 = blockIdx.x * (NV / 64) + slice;   // NV/64 = 500
    float s = 0.f;
    #pragma unroll 4
    for (int k = 0; k < 500 / 4; ++k) {
        const int v = v0 + 4 * k;
        const float x = core[(((size_t)i * NV + v) << 3) + j];
        s += x * x;
    }
    red[tid] = s;
    __syncthreads();
    if (tid < 64) {
        const float t = red[tid] + red[tid + 64] + red[tid + 128] + red[tid + 192];
        mpart[blockIdx.x * 64 + pair] = t;
    }
}

// ---------------------------------------------------------------------------
// Kernel 1b: finish M = sum_v core^2, then sequential reverse scan producing
// rights[t] (T x 8): r_{T-1}=b2n, r_t = normalize(M @ r_{t+1}); output at t is
// the carry BEFORE the step (matches lax.scan(reverse=True)).
// ---------------------------------------------------------------------------
__global__ void __launch_bounds__(64)
k_rights(const float* __restrict__ mpart, const float* __restrict__ beta,
         float* __restrict__ rights) {
    __shared__ float sM[64];
    __shared__ float r[8], nr[8];
    const int tid = threadIdx.x;  // 64 threads
    float m = 0.f;
    for (int b = 0; b < 64; ++b) m += mpart[b * 64 + tid];
    sM[tid] = m;
    if (tid < 8) { const float bb = beta[tid]; r[tid] = bb * bb; }
    __syncthreads();
    if (tid == 0) {
        float s = 0.f;
        for (int j = 0; j < 8; ++j) s += r[j];
        for (int j = 0; j < 8; ++j) r[j] /= s;
    }
    __syncthreads();
    for (int t = TT - 1; t >= 0; --t) {
        if (tid < 8) rights[t * 8 + tid] = r[tid];   // carry before step
        __syncthreads();
        if (tid < 8) {
            float acc = 0.f;
            for (int j = 0; j < 8; ++j) acc += sM[tid * 8 + j] * r[j];
            nr[tid] = acc;
        }
        __syncthreads();
        if (tid < 8) {
            float s = 0.f;
            for (int j = 0; j < 8; ++j) s += nr[j];
            r[tid] = nr[tid] / s;
        }
        __syncthreads();
    }
}

// ---------------------------------------------------------------------------
// Kernel 1c: per-batch left scan + emit GEMM A-matrix
// W[(b*T+t), i*8+j] = lefts[t,b,i] * rights[t,j].
// lefts[t] is the carry BEFORE consuming token t; l0 = alpha^2 (unnormalized,
// as in the reference); updates are sum-normalized.
// ---------------------------------------------------------------------------
__global__ void __launch_bounds__(64)
k_lefts_weights(const float* __restrict__ core, const float* __restrict__ alpha,
                const int* __restrict__ y, const float* __restrict__ rights,
                float* __restrict__ W) {
    __shared__ float l[8], ln[8];
    const int b   = blockIdx.x;
    const int tid = threadIdx.x;         // 64 threads
    const int i = tid >> 3, j = tid & 7;
    if (tid < 8) { const float a = alpha[tid]; l[tid] = a * a; }
    __syncthreads();
    for (int t = 0; t < TT; ++t) {
        // weights from the current (pre-step) left state
        W[((size_t)(b * TT + t)) * KDIM + tid] = l[i] * rights[t * 8 + j];
        const int tok = y[b * TT + t];
        if (tid < 8) {
            float acc = 0.f;
            for (int ii = 0; ii < 8; ++ii) {
                const float x = core[(((size_t)ii * NV + tok) << 3) + tid];
                acc += l[ii] * x * x;
            }
            ln[tid] = acc;
        }
        __syncthreads();                 // all reads of l[] done
        if (tid < 8) {
            float s = 0.f;
            for (int jj = 0; jj < 8; ++jj) s += ln[jj];
            l[tid] = ln[tid] / s;
        }
        __syncthreads();                 // l[] updated for next t
    }
}

// ---------------------------------------------------------------------------
// Kernel 2: p_tilde = W (2048x64) @ core^2 (64x32000) via v_wmma_f32_16x16x4.
// Block = 256 threads = 8 wave32; block tile = 16 rows x 128 cols;
// each wave owns one 16x16 output tile, K=64 -> 16 WMMA ops.
// A tile staged into LDS with the gfx1250 async path
// (global_load_async_to_lds_b128 + s_wait_asynccnt).
// Each block also emits deterministic per-row partial sums (its own slot in
// rowpart) so the normalize kernel never re-reads out to compute sums.
// Output stores are non-temporal: out streams once, core^2 stays L2-resident.
// ---------------------------------------------------------------------------
__global__ void __launch_bounds__(256)
k_gemm(const float* __restrict__ core, const float* __restrict__ W,
       float* __restrict__ out, float* __restrict__ rowpart) {
    __shared__ float Asmem[16 * 68];       // stride 68: conflict-free b64 reads
    __shared__ float ptile[16][132];       // per-block 16x128 tile copy (padded)
    __shared__ float red2[16][16];
    const int tid     = threadIdx.x;
    const int rowBase = blockIdx.y * 16;
    const int colBase = blockIdx.x * 128;

    // --- Stage A tile (16 rows x 64 K) into LDS via async copy (16B/lane) ---
    {
        const int e    = tid * 4;          // 0..1020; 4 floats stay in one row
        const int rrow = e >> 6;
        const int kk   = e & 63;
        const uint64_t gaddr = (uint64_t)(uintptr_t)
            (W + (size_t)(rowBase + rrow) * KDIM + kk);
        const uint32_t laddr = (uint32_t)(uintptr_t)&Asmem[rrow * 68 + kk];
        asm volatile("global_load_async_to_lds_b128 %0, %1, off"
                     :: "v"(laddr), "v"(gaddr) : "memory");
        asm volatile("s_wait_asynccnt 0" ::: "memory");
    }
    __syncthreads();

    const int lane   = tid & 31;
    const int wave   = tid >> 5;
    const int laneLo = lane & 15;
    const int laneHi = lane >> 4;
    const int n      = colBase + wave * 16 + laneLo;   // output column

    v8f acc = {};
    #pragma unroll
    for (int s = 0; s < 16; ++s) {
        const int kk = 4 * s + 2 * laneHi;             // K offset, always even
        // A operand: two consecutive K values for row laneLo (LDS, b64 read)
        const v2f a = *(const v2f*)&Asmem[laneLo * 68 + kk];
        // B operand: core^2 with k=i*8+j; kk even => j,j+1 contiguous in mem
        const int i = kk >> 3, j = kk & 7;
        const float* bp = core + (((size_t)i * NV + n) << 3) + j;
        const v2f braw = *(const v2f*)bp;
        v2f bsq; bsq.x = braw.x * braw.x; bsq.y = braw.y * braw.y;
        acc = __builtin_amdgcn_wmma_f32_16x16x4_f32(
            /*neg_a=*/false, a, /*neg_b=*/false, bsq,
            /*c_mod=*/(short)0, acc, /*reuse_a=*/false, /*reuse_b=*/false);
    }

    // --- Stream results (non-temporal) and drop a copy in LDS for row sums ---
    #pragma unroll
    for (int r = 0; r < 8; ++r) {
        const int rloc = r + 8 * laneHi;
        __builtin_nontemporal_store(acc[r],
            out + (size_t)(rowBase + rloc) * NV + n);
        ptile[rloc][wave * 16 + laneLo] = acc[r];
    }
    __syncthreads();

    // Deterministic fixed-order reduction of the 16x128 tile to 16 row sums.
    {
        const int row = tid >> 4;          // 0..15
        const int seg = tid & 15;          // 0..15, 8 columns each
        float s = 0.f;
        #pragma unroll
        for (int c = 0; c < 8; ++c) s += ptile[row][seg * 8 + c];
        red2[row][seg] = s;
    }
    __syncthreads();
    if (tid < 16) {
        float t = 0.f;
        #pragma unroll
        for (int s2 = 0; s2 < 16; ++s2) t += red2[tid][s2];
        rowpart[(size_t)(rowBase + tid) * NCB + blockIdx.x] = t;  // own slot
    }
}

// ---------------------------------------------------------------------------
// Kernel 3: per-row normalization. Row sum = fixed-order reduction of the 250
// per-block partials (deterministic), then a single non-temporal scale pass.
// ---------------------------------------------------------------------------
__global__ void __launch_bounds__(256)
k_normalize(float* __restrict__ out, const float* __restrict__ rowpart) {
    __shared__ float red[256];
    const int row  = blockIdx.x;
    const size_t base = (size_t)row * NV;
    const int tid = threadIdx.x;
    red[tid] = (tid < NCB) ? rowpart[(size_t)row * NCB + tid] : 0.f;
    __syncthreads();
    #pragma unroll
    for (int off = 128; off > 0; off >>= 1) {
        if (tid < off) red[tid] += red[tid + off];
        __syncthreads();
    }
    const float inv = 1.0f / red[0];
    for (int v = tid; v < NV; v += 256) {
        const float x = __builtin_nontemporal_load(out + base + v);
        __builtin_nontemporal_store(x * inv, out + base + v);
    }
}

// ---------------------------------------------------------------------------
// Launch. Workspace layout (floats):
//   [0,4096)          : M partials (64 blocks x 64)
//   [4160,4672)       : rights (64 x 8)
//   [8192,139264)     : W (2048 x 64)
//   [139264,651264)   : rowpart (2048 x 250)      (~2.6 MB total)
// ---------------------------------------------------------------------------
extern "C" void kernel_launch(void* const* d_in, const int* in_sizes, int n_in,
                              void* d_out, int out_size, void* d_ws, size_t ws_size,
                              hipStream_t stream) {
    const float* alpha = (const float*)d_in[0];
    const float* beta  = (const float*)d_in[1];
    const float* core  = (const float*)d_in[2];
    const int*   y     = (const int*)d_in[3];
    float* out = (float*)d_out;
    float* ws  = (float*)d_ws;

    float* mpart   = ws;           // 4096 floats
    float* rights  = ws + 4160;    // 512 floats
    float* W       = ws + 8192;    // 131072 floats
    float* rowpart = ws + 139264;  // 512000 floats

    k_marg_partial<<<64, 256, 0, stream>>>(core, mpart);
    k_rights<<<1, 64, 0, stream>>>(mpart, beta, rights);
    k_lefts_weights<<<BB, 64, 0, stream>>>(core, alpha, y, rights, W);

    dim3 g2(NCB, NROWS / 16);      // 250 x 128 blocks
    k_gemm<<<g2, 256, 0, stream>>>(core, W, out, rowpart);

    k_normalize<<<NROWS, 256, 0, stream>>>(out, rowpart);
}